// SimpleGCN_79568564125765
// MI455X (gfx1250) — compile-verified
//
#include <hip/hip_runtime.h>
#include <math.h>

// ---- problem constants (match reference) ----
#define BG   64
#define PERG 512
#define NG   32768
#define EG   1048576
#define FG   128
#define NFG  (NG * FG)
#define QBLOCKS 1024   // qloss blocks per call (64 graphs * 16 row-strips)

typedef __attribute__((ext_vector_type(2))) float v2f;
typedef __attribute__((ext_vector_type(8))) float v8f;
typedef __attribute__((ext_vector_type(4))) unsigned uint4v;
typedef __attribute__((ext_vector_type(8))) int int8v;
typedef __attribute__((ext_vector_type(4))) int int4v;

__device__ __forceinline__ v8f wmma4(v2f a, v2f b, v8f c) {
  // V_WMMA_F32_16X16X4_F32 : D = A(16x4) * B(4x16) + C  (fp32 native matrix op)
  return __builtin_amdgcn_wmma_f32_16x16x4_f32(false, a, false, b, (short)0, c, false, false);
}

#define HAVE_TDM __has_builtin(__builtin_amdgcn_tensor_load_to_lds)

#if HAVE_TDM
// raw LDS byte offset of a __shared__ object (AS3 pointers are ds-instruction offsets)
__device__ __forceinline__ unsigned lds_offset(const void* p) {
  return (unsigned)(unsigned long long)(const __attribute__((address_space(3))) char*)p;
}

// Issue one 2-D TDM tile load: tile_d0 x tile_d1 elements (4B each), row stride
// stride0 elements, from gaddr into LDS at lds_addr.  pad_bits: pre-shifted
// pad_enable/pad_interval/pad_amount fields for group1 dword0 (or 0).
__device__ __forceinline__ void tdm_load_2d(unsigned lds_addr, const void* gaddr,
                                            unsigned tensor_d0, unsigned tensor_d1,
                                            unsigned tile_d0, unsigned tile_d1,
                                            unsigned stride0, unsigned pad_bits) {
  unsigned long long ga = (unsigned long long)gaddr;
  uint4v g0;
  g0[0] = 1u;                                                  // count=1, user mode
  g0[1] = lds_addr;                                            // LDS byte address
  g0[2] = (unsigned)ga;                                        // global_addr[31:0]
  g0[3] = (unsigned)((ga >> 32) & 0x01FFFFFFu) | 0x80000000u;  // addr[56:32] | type=2
  int8v g1;
  g1[0] = (int)((2u << 16) | pad_bits);                        // data_size=4B (+pad cfg)
  g1[1] = (int)((tensor_d0 & 0xFFFFu) << 16);                  // abar=0 | td0.lo16
  g1[2] = (int)((tensor_d0 >> 16) | ((tensor_d1 & 0xFFFFu) << 16));
  g1[3] = (int)((tensor_d1 >> 16) | (tile_d0 << 16));
  g1[4] = (int)(tile_d1 & 0xFFFFu);                            // tile_dim1 | tile_dim2=0
  g1[5] = (int)stride0;                                        // dim0 stride lo32
  g1[6] = 0;                                                   // stride0.hi16 | stride1.lo16
  g1[7] = 0;                                                   // stride1.hi32
  int4v z4 = {0, 0, 0, 0};
  int8v z8 = {0, 0, 0, 0, 0, 0, 0, 0};
  // 6-arg (clang-23 / therock) form: groups 0..3 + extra group + cache policy
  __builtin_amdgcn_tensor_load_to_lds(g0, g1, z4, z4, z8, 0);
}
#endif

// ---------------- threefry2x32 (dropout RNG) ----------------
__device__ __forceinline__ unsigned rotl32(unsigned x, int r) { return (x << r) | (x >> (32 - r)); }

__device__ __forceinline__ void threefry2x32(unsigned k0, unsigned k1, unsigned c0, unsigned c1,
                                             unsigned& o0, unsigned& o1) {
  const unsigned ks2 = k0 ^ k1 ^ 0x1BD11BDAu;
  unsigned x0 = c0 + k0, x1 = c1 + k1;
  const int RA[4] = {13, 15, 26, 6};
  const int RB[4] = {17, 29, 16, 24};
#pragma unroll
  for (int i = 0; i < 4; ++i) { x0 += x1; x1 = rotl32(x1, RA[i]); x1 ^= x0; }
  x0 += k1; x1 += ks2 + 1;
#pragma unroll
  for (int i = 0; i < 4; ++i) { x0 += x1; x1 = rotl32(x1, RB[i]); x1 ^= x0; }
  x0 += ks2; x1 += k0 + 2;
#pragma unroll
  for (int i = 0; i < 4; ++i) { x0 += x1; x1 = rotl32(x1, RA[i]); x1 ^= x0; }
  x0 += k0; x1 += k1 + 3;
#pragma unroll
  for (int i = 0; i < 4; ++i) { x0 += x1; x1 = rotl32(x1, RB[i]); x1 ^= x0; }
  x0 += k1; x1 += ks2 + 4;
#pragma unroll
  for (int i = 0; i < 4; ++i) { x0 += x1; x1 = rotl32(x1, RA[i]); x1 ^= x0; }
  o0 = x0 + ks2; o1 = x1 + k0 + 5;
}

// ---------------- graph preprocessing ----------------
__global__ void zero_deg_kernel(int* __restrict__ deg) {
  int i = blockIdx.x * blockDim.x + threadIdx.x;
  if (i < NG) deg[i] = 0;
}

__global__ void count_edges_kernel(const int* __restrict__ dst, int* __restrict__ deg) {
  int e = blockIdx.x * blockDim.x + threadIdx.x;
  if (e < EG) atomicAdd(&deg[dst[e]], 1);
}

__global__ void scan_kernel(const int* __restrict__ deg, int* __restrict__ rowptr,
                            int* __restrict__ cursor) {
  __shared__ int s[1024];
  const int t = threadIdx.x;
  const int base = t * 32;
  int local[32];
  int sum = 0;
#pragma unroll
  for (int j = 0; j < 32; ++j) { local[j] = deg[base + j]; sum += local[j]; }
  s[t] = sum;
  __syncthreads();
  for (int off = 1; off < 1024; off <<= 1) {
    int v = (t >= off) ? s[t - off] : 0;
    __syncthreads();
    s[t] += v;
    __syncthreads();
  }
  int run = s[t] - sum;  // exclusive prefix for this chunk
#pragma unroll
  for (int j = 0; j < 32; ++j) {
    rowptr[base + j] = run;
    cursor[base + j] = run;
    run += local[j];
  }
  if (t == 1023) rowptr[NG] = run;
}

__global__ void fill_csr_kernel(const int* __restrict__ dst, int* __restrict__ cursor,
                                int* __restrict__ eids) {
  int e = blockIdx.x * blockDim.x + threadIdx.x;
  if (e < EG) {
    int p = atomicAdd(&cursor[dst[e]], 1);
    eids[p] = e;
  }
}

// canonical order within each bucket -> deterministic float sums later
__global__ void sort_lists_kernel(const int* __restrict__ rowptr, int* __restrict__ eids) {
  int n = blockIdx.x * blockDim.x + threadIdx.x;
  if (n >= NG) return;
  int lo = rowptr[n], hi = rowptr[n + 1];
  for (int i = lo + 1; i < hi; ++i) {
    int v = eids[i];
    int j = i - 1;
    while (j >= lo && eids[j] > v) { eids[j + 1] = eids[j]; --j; }
    eids[j + 1] = v;
  }
}

// ---------------- qloss : sum_b sum_{n,f} X_b[n,f] * (net @ X_b)[n,f] ----------------
// grid = 64 graphs * 16 row-strips, 256 threads (8 waves).
// TDM double-buffered staging (net tile padded to stride 36 by the DMA engine)
// overlapped with fp32 WMMA; fallback to synchronous float4 staging if no TDM builtin.
#define NSTR 36  // LDS row stride for net tile (bank-conflict-free A-fragment reads)
__global__ __launch_bounds__(256) void qloss_kernel(const float* __restrict__ x,
                                                    const float* __restrict__ net,
                                                    float* __restrict__ partials, int slot) {
  __shared__ float netlds[2][32 * NSTR];
  __shared__ float xlds[2][32 * 128];
  __shared__ float warr[8];

  const int b = blockIdx.x >> 4;
  const int rowbase = (blockIdx.x & 15) * 32;
  const float* __restrict__ Xb = x + (size_t)b * PERG * FG;
  const float* __restrict__ netR = net + (size_t)rowbase * 512;

  const int t = threadIdx.x;
  const int w = t >> 5, lane = t & 31;
  const int half = lane >> 4, l16 = lane & 15;
  const int ct16 = w * 16;

  v8f acc0 = {};
  v8f acc1 = {};

#if HAVE_TDM
  const unsigned padbits = (1u << 20) | (4u << 22) | (3u << 25);  // pad every 32 dw by 4 dw
  const unsigned nloff0 = lds_offset(&netlds[0][0]);
  const unsigned nloff1 = lds_offset(&netlds[1][0]);
  const unsigned xloff0 = lds_offset(&xlds[0][0]);
  const unsigned xloff1 = lds_offset(&xlds[1][0]);
  // prologue: DMA tiles for m0=0 into buffer 0 (one wave issues, TDM ignores EXEC)
  if (w == 0) {
    tdm_load_2d(nloff0, netR, 32u, 32u, 32u, 32u, 512u, padbits);
    tdm_load_2d(xloff0, Xb, 128u, 32u, 128u, 32u, 128u, 0u);
  }
#endif

  for (int i = 0; i < 16; ++i) {
    const int m0 = i * 32;
    const int cur = i & 1;
#if HAVE_TDM
    if (w == 0) {
      if (m0 + 32 < 512) {  // DMA next tiles into the other buffer, overlap with compute
        const unsigned nl = (cur ? nloff0 : nloff1);
        const unsigned xl = (cur ? xloff0 : xloff1);
        tdm_load_2d(nl, netR + (m0 + 32), 32u, 32u, 32u, 32u, 512u, padbits);
        tdm_load_2d(xl, Xb + (size_t)(m0 + 32) * 128, 128u, 32u, 128u, 32u, 128u, 0u);
        __builtin_amdgcn_s_wait_tensorcnt(2);  // current buffer's 2 DMAs complete
      } else {
        __builtin_amdgcn_s_wait_tensorcnt(0);
      }
    }
    __syncthreads();  // release all waves: current buffer ready
#else
    // synchronous staging fallback
    {
      int r = t >> 3, c4 = (t & 7) * 4;
      *(float4*)(&netlds[cur][r * NSTR + c4]) =
          *(const float4*)(netR + (size_t)r * 512 + m0 + c4);
    }
#pragma unroll
    for (int j = 0; j < 4; ++j) {
      int idx = t + j * 256;
      int r = idx >> 5, c4 = (idx & 31) * 4;
      *(float4*)(&xlds[cur][r * 128 + c4]) = *(const float4*)(Xb + (size_t)(m0 + r) * 128 + c4);
    }
    if (m0 + 32 < 512) {
      int r = t >> 3;
      __builtin_prefetch(Xb + (size_t)(m0 + 32 + r) * 128 + (t & 7) * 16, 0, 3);
    }
    __syncthreads();
#endif

    const float* __restrict__ nl = netlds[cur];
    const float* __restrict__ xl = xlds[cur];
#pragma unroll
    for (int k = 0; k < 32; k += 4) {
      v2f bf;
      bf.x = xl[(k + 2 * half) * 128 + ct16 + l16];
      bf.y = xl[(k + 2 * half + 1) * 128 + ct16 + l16];
      v2f a0;
      a0.x = nl[l16 * NSTR + k + 2 * half];
      a0.y = nl[l16 * NSTR + k + 2 * half + 1];
      v2f a1;
      a1.x = nl[(16 + l16) * NSTR + k + 2 * half];
      a1.y = nl[(16 + l16) * NSTR + k + 2 * half + 1];
      acc0 = wmma4(a0, bf, acc0);
      acc1 = wmma4(a1, bf, acc1);
    }
    __syncthreads();  // all waves done with current buffer before it is re-filled
  }

  // dot C-tiles with X rows (trace accumulation)
  float p = 0.f;
#pragma unroll
  for (int r = 0; r < 8; ++r) {
    int n0 = rowbase + r + 8 * half;
    p += acc0[r] * Xb[(size_t)n0 * 128 + ct16 + l16];
    p += acc1[r] * Xb[(size_t)(n0 + 16) * 128 + ct16 + l16];
  }
#pragma unroll
  for (int o = 16; o > 0; o >>= 1) p += __shfl_down(p, o);
  if (lane == 0) warr[w] = p;
  __syncthreads();
  if (t == 0) {
    float s = 0.f;
#pragma unroll
    for (int i = 0; i < 8; ++i) s += warr[i];
    partials[slot * QBLOCKS + blockIdx.x] = s;
  }
}

// ---------------- dropout (p=0.5, scale x2) ----------------
__global__ void dropout_kernel(const float* __restrict__ in, float* __restrict__ out,
                               unsigned layer) {
  int i = blockIdx.x * blockDim.x + threadIdx.x;
  unsigned r0, r1;
  threefry2x32(0x2Au, 0x9E3779B9u + layer, (unsigned)i, 0x6A09E667u, r0, r1);
  float v = in[i];
  out[i] = (r0 >> 31) ? 0.0f : 2.0f * v;
}

// ---------------- weighted-mean aggregation (wave per node, CSR order) ----------------
__global__ __launch_bounds__(256) void aggregate_kernel(const float* __restrict__ in,
                                                        const int* __restrict__ src,
                                                        const float* __restrict__ ew,
                                                        const int* __restrict__ rowptr,
                                                        const int* __restrict__ eids,
                                                        float* __restrict__ out) {
  const int n = blockIdx.x * 8 + (threadIdx.x >> 5);
  const int lane = threadIdx.x & 31;
  const int f0 = lane * 4;
  const float4 sv = *(const float4*)(in + (size_t)n * FG + f0);  // self loop, weight 1
  float ax = sv.x, ay = sv.y, az = sv.z, aw = sv.w;
  const int lo = rowptr[n], hi = rowptr[n + 1];
  for (int idx = lo; idx < hi; ++idx) {
    int e = eids[idx];
    int s = src[e];
    float wgt = ew[e];
    const float4 xv = *(const float4*)(in + (size_t)s * FG + f0);
    ax += wgt * xv.x; ay += wgt * xv.y; az += wgt * xv.z; aw += wgt * xv.w;
  }
  const float inv = 1.0f / (float)(hi - lo + 1);
  float4 ov;
  ov.x = ax * inv; ov.y = ay * inv; ov.z = az * inv; ov.w = aw * inv;
  *(float4*)(out + (size_t)n * FG + f0) = ov;
}

// ---------------- linear + bias + L2-normalize + leaky ReLU (WMMA fp32) ----------------
// block = 256 threads (8 waves), handles 64 rows x 128 cols.  K = 128.
#define ASTR 132  // 132*4B = 528B rows: 16B aligned, conflict-free for stride-132 lane access
__global__ __launch_bounds__(256) void linear_norm_kernel(const float* __restrict__ in,
                                                          const float* __restrict__ W,
                                                          const float* __restrict__ bias,
                                                          float* __restrict__ out) {
  extern __shared__ float smem[];
  float* Ablk = smem;                    // 64*132 floats
  float* Wlds = smem + 64 * ASTR;        // 128*128 floats
  float* nrm  = Wlds + 128 * 128;        // 256 floats
  float* outbuf = Ablk;                  // alias after compute (stride 128)

  const int row0 = blockIdx.x * 64;
  const int t = threadIdx.x;
  const int w = t >> 5, lane = t & 31;
  const int half = lane >> 4, l16 = lane & 15;
  const int rt = w >> 1;
  const int ctbase = (w & 1) * 4;

  // stage A block (64x128)
#pragma unroll
  for (int j = 0; j < 8; ++j) {
    int idx = t + j * 256;
    int r = idx >> 5, c4 = (idx & 31) * 4;
    *(float4*)(Ablk + r * ASTR + c4) = *(const float4*)(in + (size_t)(row0 + r) * FG + c4);
  }
  // stage full W (128x128)
#pragma unroll
  for (int j = 0; j < 16; ++j) {
    int idx = t + j * 256;
    int r = idx >> 5, c4 = (idx & 31) * 4;
    *(float4*)(Wlds + r * 128 + c4) = *(const float4*)(W + (size_t)r * FG + c4);
  }
  __syncthreads();

  v8f acc[4] = {};
#pragma unroll
  for (int k = 0; k < 128; k += 4) {
    v2f a;
    a.x = Ablk[(rt * 16 + l16) * ASTR + k + 2 * half];
    a.y = Ablk[(rt * 16 + l16) * ASTR + k + 2 * half + 1];
#pragma unroll
    for (int j = 0; j < 4; ++j) {
      int cc = (ctbase + j) * 16 + l16;
      v2f bf;
      bf.x = Wlds[(k + 2 * half) * 128 + cc];
      bf.y = Wlds[(k + 2 * half + 1) * 128 + cc];
      acc[j] = wmma4(a, bf, acc[j]);
    }
  }
  __syncthreads();  // everyone done reading A before aliasing it as outbuf

#pragma unroll
  for (int j = 0; j < 4; ++j) {
    int col = (ctbase + j) * 16 + l16;
    float bc = bias[col];
#pragma unroll
    for (int r = 0; r < 8; ++r) {
      int row = rt * 16 + r + 8 * half;
      outbuf[row * 128 + col] = acc[j][r] + bc;
    }
  }
  __syncthreads();

  // deterministic L2 row-norm: 4 threads per row, fixed combine order
  const int row = t >> 2, q = t & 3;
  float ss = 0.f;
#pragma unroll
  for (int c = q * 32; c < q * 32 + 32; ++c) {
    float v = outbuf[row * 128 + c];
    ss += v * v;
  }
  nrm[row * 4 + q] = ss;
  __syncthreads();
  float tot = nrm[row * 4 + 0] + nrm[row * 4 + 1] + nrm[row * 4 + 2] + nrm[row * 4 + 3];
  float inv = 1.0f / fmaxf(sqrtf(tot), 1e-12f);
#pragma unroll
  for (int c = q * 32; c < q * 32 + 32; ++c) {
    float v = outbuf[row * 128 + c] * inv;
    v = (v > 0.f) ? v : 0.01f * v;  // leaky_relu(0.01)
    out[(size_t)(row0 + row) * FG + c] = v;
  }
}

// ---------------- max pool over nodes per graph ----------------
__global__ void maxpool_kernel(const float* __restrict__ in, float* __restrict__ pooled) {
  const int b = blockIdx.x;
  const int f = threadIdx.x;
  float m = -3.402823466e+38f;
  for (int i = 0; i < PERG; ++i)
    m = fmaxf(m, in[((size_t)b * PERG + i) * FG + f]);
  pooled[b * FG + f] = m;
}

// ---------------- MLP head + deterministic loss reduce ----------------
__global__ void mlp_head_kernel(const float* __restrict__ pooled, const float* __restrict__ Wd0,
                                const float* __restrict__ bd0, const float* __restrict__ Wd1,
                                const float* __restrict__ bd1, float* __restrict__ h1g,
                                const float* __restrict__ partials, float* __restrict__ out) {
  const int t = threadIdx.x;
  // h1 = relu(pooled @ Wd0 + bd0)  : 64x256
  for (int o = t; o < 64 * 256; o += 256) {
    int bg = o >> 8, j = o & 255;
    float s = bd0[j];
    const float* hp = pooled + bg * 128;
    for (int k = 0; k < 128; ++k) s += hp[k] * Wd0[k * 256 + j];
    h1g[o] = fmaxf(s, 0.f);
  }
  __threadfence();
  __syncthreads();
  // out = h1 @ Wd1 + bd1 : 64x64
  for (int o = t; o < 64 * 64; o += 256) {
    int bg = o >> 6, j = o & 63;
    float s = bd1[j];
    const float* hp = h1g + bg * 256;
    for (int k = 0; k < 256; ++k) s += hp[k] * Wd1[k * 64 + j];
    out[o] = s;
  }
  // loss = sum of all 4*1024 qloss partials (fixed-order tree -> deterministic)
  __shared__ float red[256];
  float s = 0.f;
  for (int i = t * 16; i < t * 16 + 16; ++i) s += partials[i];
  red[t] = s;
  __syncthreads();
  for (int o = 128; o > 0; o >>= 1) {
    if (t < o) red[t] += red[t + o];
    __syncthreads();
  }
  if (t == 0) out[4096] = red[0];
}

// ---------------- host-side orchestration ----------------
extern "C" void kernel_launch(void* const* d_in, const int* in_sizes, int n_in,
                              void* d_out, int out_size, void* d_ws, size_t ws_size,
                              hipStream_t stream) {
  (void)in_sizes; (void)n_in; (void)out_size; (void)ws_size;
  const float* x    = (const float*)d_in[0];
  const int*   ei   = (const int*)d_in[1];
  const float* ea   = (const float*)d_in[2];
  const float* net  = (const float*)d_in[5];
  const float* W0   = (const float*)d_in[6];
  const float* b0   = (const float*)d_in[7];
  const float* W1   = (const float*)d_in[8];
  const float* b1   = (const float*)d_in[9];
  const float* W2   = (const float*)d_in[10];
  const float* b2   = (const float*)d_in[11];
  const float* Wd0  = (const float*)d_in[12];
  const float* bd0  = (const float*)d_in[13];
  const float* Wd1  = (const float*)d_in[14];
  const float* bd1  = (const float*)d_in[15];
  const int* srcArr = ei;
  const int* dstArr = ei + EG;
  float* out = (float*)d_out;

  // workspace layout
  float* ws       = (float*)d_ws;
  float* bufA     = ws;
  float* bufB     = bufA + NFG;
  float* bufC     = bufB + NFG;
  float* pooled   = bufC + NFG;          // 64*128
  float* h1g      = pooled + 64 * 128;   // 64*256
  float* partials = h1g + 64 * 256;      // 4*1024
  int* deg    = (int*)(partials + 4 * QBLOCKS);
  int* rowptr = deg + NG;                // NG+1
  int* cursor = rowptr + NG + 1;
  int* eids   = cursor + NG;             // EG

  const size_t linShmem = (size_t)(64 * ASTR + 128 * 128 + 256) * sizeof(float);

  // x -> bufA (do not mutate inputs)
  (void)hipMemcpyAsync(bufA, x, (size_t)NFG * sizeof(float), hipMemcpyDeviceToDevice, stream);

  // CSR build (deterministic after sort)
  zero_deg_kernel<<<NG / 256, 256, 0, stream>>>(deg);
  count_edges_kernel<<<EG / 256, 256, 0, stream>>>(dstArr, deg);
  scan_kernel<<<1, 1024, 0, stream>>>(deg, rowptr, cursor);
  fill_csr_kernel<<<EG / 256, 256, 0, stream>>>(dstArr, cursor, eids);
  sort_lists_kernel<<<NG / 128, 128, 0, stream>>>(rowptr, eids);

  // loss += qloss(x)
  qloss_kernel<<<QBLOCKS, 256, 0, stream>>>(bufA, net, partials, 0);

  // layer 0
  aggregate_kernel<<<NG / 8, 256, 0, stream>>>(bufA, srcArr, ea, rowptr, eids, bufB);
  linear_norm_kernel<<<NG / 64, 256, linShmem, stream>>>(bufB, W0, b0, bufC);
  qloss_kernel<<<QBLOCKS, 256, 0, stream>>>(bufC, net, partials, 1);

  // layer 1 (dropout first)
  dropout_kernel<<<NFG / 256, 256, 0, stream>>>(bufC, bufA, 1u);
  aggregate_kernel<<<NG / 8, 256, 0, stream>>>(bufA, srcArr, ea, rowptr, eids, bufB);
  linear_norm_kernel<<<NG / 64, 256, linShmem, stream>>>(bufB, W1, b1, bufA);
  qloss_kernel<<<QBLOCKS, 256, 0, stream>>>(bufA, net, partials, 2);

  // layer 2 (dropout first)
  dropout_kernel<<<NFG / 256, 256, 0, stream>>>(bufA, bufC, 2u);
  aggregate_kernel<<<NG / 8, 256, 0, stream>>>(bufC, srcArr, ea, rowptr, eids, bufB);
  linear_norm_kernel<<<NG / 64, 256, linShmem, stream>>>(bufB, W2, b2, bufC);
  qloss_kernel<<<QBLOCKS, 256, 0, stream>>>(bufC, net, partials, 3);

  // head
  maxpool_kernel<<<BG, FG, 0, stream>>>(bufC, pooled);
  mlp_head_kernel<<<1, 256, 0, stream>>>(pooled, Wd0, bd0, Wd1, bd1, h1g, partials, out);
}